// NKRRSolver_14731737825437
// MI455X (gfx1250) — compile-verified
//
#include <hip/hip_runtime.h>

typedef __attribute__((ext_vector_type(2))) float v2f;
typedef __attribute__((ext_vector_type(8))) float v8f;

#define DIM    32
#define TILE   16
#define MCONST 4096
#define LOG2E  1.4426950408889634f

// B fragment for one 16-center tile: B[k][n] = C[jt+n][k].
// Lane holds n = ln, k = 4*kk + koff{,+1} (32-bit WMMA B layout, wave32).
__device__ __forceinline__ void load_bfrag(v2f b[8], const float* __restrict__ crow,
                                           int koff) {
    #pragma unroll
    for (int kk = 0; kk < 8; ++kk) {
        b[kk].x = crow[4 * kk + koff + 0];
        b[kk].y = crow[4 * kk + koff + 1];
    }
}

// Chain 8x V_WMMA_F32_16X16X4_F32 to cover D=32 in full f32 precision.
__device__ __forceinline__ v8f wmma_chain(const v2f a[8], const v2f b[8]) {
    v8f d = {};
    #pragma unroll
    for (int kk = 0; kk < 8; ++kk) {
        // 8 args: (neg_a, A, neg_b, B, c_mod, C, reuse_a, reuse_b)
        d = __builtin_amdgcn_wmma_f32_16x16x4_f32(
            false, a[kk], false, b[kk], (short)0, d, false, false);
    }
    return d;
}

// exp(-0.5*sqdist/g^2)*alpha accumulate, folded to base-2:
//   e = c1*x2 + c1*y2 + c2*dot, c2 = -2*c1, c1 = -0.5*log2e/g^2
// -> per element: v_add + v_fma + v_exp_f32 + v_fma (3 VALU + 1 TRANS).
__device__ __forceinline__ void epilogue(const v8f d, float y2e, float al, float c2,
                                         const float x2e[8], float acc[8]) {
    #pragma unroll
    for (int v = 0; v < 8; ++v) {
        float e = __builtin_fmaf(c2, d[v], x2e[v] + y2e);
        acc[v] = __builtin_fmaf(al, __builtin_amdgcn_exp2f(e), acc[v]);
    }
}

__global__ __launch_bounds__(256, 2)
void nkrr_rbf_wmma_kernel(const float* __restrict__ X,
                          const float* __restrict__ C,
                          const float* __restrict__ alphas,
                          const float* __restrict__ sigma,
                          float* __restrict__ out,
                          int N, int M)
{
    __shared__ float s_y2e[MCONST];   // c1 * ||c_j||^2
    __shared__ float s_al[MCONST];

    const float g  = sigma[0];
    const float c1 = -0.5f * LOG2E / (g * g);
    const float c2 = -2.0f * c1;

    // Workgroup prologue: scaled center norms + alphas into LDS (32 KB).
    for (int j = threadIdx.x; j < M; j += blockDim.x) {
        const float* c = C + (size_t)j * DIM;
        float s = 0.f;
        #pragma unroll
        for (int d = 0; d < DIM; ++d) s += c[d] * c[d];
        s_y2e[j] = c1 * s;
        s_al[j]  = alphas[j];
    }
    __syncthreads();

    const int lane = threadIdx.x & 31;
    const int wave = threadIdx.x >> 5;
    const int wavesPerBlock = blockDim.x >> 5;
    const int rowTile = blockIdx.x * wavesPerBlock + wave;
    const int row0 = rowTile * TILE;
    if (row0 >= N) return;

    const int half = lane >> 4;   // 0: lanes 0-15, 1: lanes 16-31
    const int ln   = lane & 15;
    const int koff = half * 2;    // K-pair selector per 32-bit A/B fragment layout

    // A fragments: 16x4 f32 per k-step. Lane holds X[row0+ln, 4*kk+koff .. +1].
    v2f a[8];
    {
        const float* xrow = X + (size_t)(row0 + ln) * DIM;
        #pragma unroll
        for (int kk = 0; kk < 8; ++kk) {
            a[kk].x = xrow[4 * kk + koff + 0];
            a[kk].y = xrow[4 * kk + koff + 1];
        }
    }

    // Pre-scaled row norms for the rows this lane's C/D fragment covers:
    // m = v + 8*half, v = 0..7.
    float x2e[8];
    #pragma unroll
    for (int v = 0; v < 8; ++v) {
        const float* xr = X + (size_t)(row0 + v + 8 * half) * DIM;
        float s = 0.f;
        #pragma unroll
        for (int d = 0; d < DIM; ++d) s += xr[d] * xr[d];
        x2e[v] = c1 * s;
    }

    float acc[8];
    #pragma unroll
    for (int v = 0; v < 8; ++v) acc[v] = 0.f;

    // Software-pipelined sweep over M in 16-center tiles, 2 tiles/iteration,
    // double-buffered B fragments so VMEM overlaps the WMMA + exp streams.
    const float* cbase = C + (size_t)ln * DIM;   // row `ln` of tile 0

    v2f b0[8], b1[8];
    load_bfrag(b0, cbase, koff);                 // prefetch tile 0

    for (int jt = 0; jt < M; jt += 2 * TILE) {
        // prefetch tile jt+16 while tile jt computes
        load_bfrag(b1, cbase + (size_t)(jt + TILE) * DIM, koff);

        v8f d0 = wmma_chain(a, b0);
        epilogue(d0, s_y2e[jt + ln], s_al[jt + ln], c2, x2e, acc);

        // prefetch tile jt+32 (clamped in-bounds; values unused on last iter)
        int jn = jt + 2 * TILE;
        if (jn >= M) jn = 0;
        load_bfrag(b0, cbase + (size_t)jn * DIM, koff);

        v8f d1 = wmma_chain(a, b1);
        epilogue(d1, s_y2e[jt + TILE + ln], s_al[jt + TILE + ln], c2, x2e, acc);
    }

    // Reduce over the 16 columns held across lanes of each 16-lane half
    // (xor masks 8/4/2/1 keep bit4 fixed -> stay within the half). wave32.
    #pragma unroll
    for (int v = 0; v < 8; ++v) {
        float s = acc[v];
        s += __shfl_xor(s, 8, 32);
        s += __shfl_xor(s, 4, 32);
        s += __shfl_xor(s, 2, 32);
        s += __shfl_xor(s, 1, 32);
        if (ln == 0) {
            out[row0 + v + 8 * half] = s;
        }
    }
}

extern "C" void kernel_launch(void* const* d_in, const int* in_sizes, int n_in,
                              void* d_out, int out_size, void* d_ws, size_t ws_size,
                              hipStream_t stream) {
    const float* X      = (const float*)d_in[0];
    const float* C      = (const float*)d_in[1];
    const float* alphas = (const float*)d_in[2];
    const float* sigma  = (const float*)d_in[3];
    float* out = (float*)d_out;

    const int N = in_sizes[0] / DIM;   // 32768
    const int M = in_sizes[1] / DIM;   // 4096

    const int wavesPerBlock = 8;                 // 256 threads = 8 wave32
    const int rowTiles = N / TILE;               // 2048 waves
    dim3 block(32 * wavesPerBlock);
    dim3 grid(rowTiles / wavesPerBlock);         // 256 workgroups

    nkrr_rbf_wmma_kernel<<<grid, block, 0, stream>>>(X, C, alphas, sigma, out, N, M);
}